// SDE_79809082294796
// MI455X (gfx1250) — compile-verified
//
#include <hip/hip_runtime.h>
#include <math.h>

// Problem constants (match reference: BATCH=32768, L=1024, RHO=0.5, SIGMA=0.3)
#define BATCH_C 32768
#define L_C     1024
#define TPB     64     // threads per block == paths per block (2 waves)
#define TT      32     // time-tile length

// Native clang vector types (HIP's float2/int4 wrapper classes are not
// accepted by the amdgcn builtins).
typedef int   v4i __attribute__((ext_vector_type(4)));
typedef float v2f __attribute__((ext_vector_type(2)));

// Address-space-qualified pointer types for the gfx1250 async-LDS builtin:
// signature (per clang diagnostic) takes v4i AS(1)* and v4i AS(3)*.
typedef __attribute__((address_space(1))) v4i gv4i;
typedef __attribute__((address_space(3))) v4i lv4i;

// One 16-byte async copy global -> LDS (per-lane). Falls back to a plain
// register copy if the builtin is not declared (e.g. host pass).
__device__ __forceinline__ void async_copy16(const void* gsrc, void* ldst) {
#if __has_builtin(__builtin_amdgcn_global_load_async_to_lds_b128)
  __builtin_amdgcn_global_load_async_to_lds_b128(
      (gv4i*)(void*)gsrc, (lv4i*)ldst, /*offset=*/0, /*cpol=*/0);
#else
  *(v4i*)ldst = *(const v4i*)gsrc;
#endif
}

__device__ __forceinline__ void wait_async0() {
#if __has_builtin(__builtin_amdgcn_s_wait_asynccnt)
  __builtin_amdgcn_s_wait_asynccnt(0);
#elif defined(__gfx1250__)
  asm volatile("s_wait_asynccnt 0" ::: "memory");
#endif
}

__global__ __launch_bounds__(TPB) void sde_path_kernel(
    const float* __restrict__ x0, const float* __restrict__ y0,
    const float* __restrict__ ts, const float* __restrict__ z,
    float* __restrict__ out)
{
  // LDS: 32KB z double-buffer + 16KB output tile + 8KB step sizes = 56KB
  __shared__ float zbuf[2][TT][TPB][2];
  __shared__ float obuf[TPB][TT][2];
  __shared__ float s_h [L_C];
  __shared__ float s_sh[L_C];

  const int tid   = threadIdx.x;
  const int pBase = blockIdx.x * TPB;
  const int path  = pBase + tid;

  // Step sizes h and sqrt(h), computed once per block into LDS.
  for (int i = tid; i < L_C - 1; i += TPB) {
    float h = ts[i + 1] - ts[i];
    s_h[i]  = h;
    s_sh[i] = sqrtf(h);
  }

  float x = x0[path];
  float y = y0[path];

  // t = 0 trajectory entry (one scattered 8B store per path; negligible).
  {
    v2f v; v.x = x; v.y = y;
    *((v2f*)out + (size_t)path * L_C) = v;
  }

  const int NSTEP = L_C - 1;                 // 1023
  const int NT    = (NSTEP + TT - 1) / TT;   // 32 tiles (31 full + one of 31)

  // Stage tile 0 into zbuf[0]: rows = steps, 64 paths * float2 = 512B/row,
  // 32 x 16B segments per row, flat-assigned across the 64 threads.
  {
    const int len = (TT < NSTEP) ? TT : NSTEP;
    const char* zb = (const char*)z + (size_t)pBase * 8;
    for (int s = tid; s < len * 32; s += TPB) {
      int row = s >> 5, col = s & 31;
      async_copy16(zb + (size_t)row * (BATCH_C * 8) + (size_t)col * 16,
                   &zbuf[0][row][col * 2][0]);
    }
  }
  wait_async0();
  __syncthreads();

  for (int tile = 0; tile < NT; ++tile) {
    const int t0  = tile * TT;
    const int rem = NSTEP - t0;
    const int len = (TT < rem) ? TT : rem;
    const int cur = tile & 1;

    // Kick off async prefetch of the NEXT tile into the other buffer.
    if (tile + 1 < NT) {
      const int t1   = t0 + TT;
      const int rem1 = NSTEP - t1;
      const int len1 = (TT < rem1) ? TT : rem1;
      const char* zb = (const char*)z + ((size_t)t1 * BATCH_C + pBase) * 8;
      for (int s = tid; s < len1 * 32; s += TPB) {
        int row = s >> 5, col = s & 31;
        async_copy16(zb + (size_t)row * (BATCH_C * 8) + (size_t)col * 16,
                     &zbuf[cur ^ 1][row][col * 2][0]);
      }
    }

    // Serial recurrence over this tile (z comes from LDS, results to LDS).
    // dW1 = sqrt(h)*z1 ; dW0 = 0.25*(sqrt(h)*z0) + 0.75*dW1
    // x' = x - x*h + 0.3*dW0 ; y' = y + x'*h + dW1
    #pragma unroll 4
    for (int j = 0; j < len; ++j) {
      const int t = t0 + j;
      float z0 = zbuf[cur][j][tid][0];
      float z1 = zbuf[cur][j][tid][1];
      float h  = s_h[t];
      float sh = s_sh[t];
      float dW1 = sh * z1;
      float dW0 = fmaf(0.25f, sh * z0, 0.75f * dW1);
      x = fmaf(-h, x, fmaf(0.3f, dW0, x));
      y = fmaf(h, x, y + dW1);
      obuf[tid][j][0] = x;
      obuf[tid][j][1] = y;
    }

    __syncthreads();  // obuf complete; zbuf[cur] fully consumed

    // Coalesced transpose-store: consecutive lanes write consecutive time
    // steps of one path's contiguous out chunk (out[(pBase+pl)*L + t0+1+j]).
    const int total = TPB * len;
    for (int f = tid; f < total; f += TPB) {
      int pl, j;
      if (len == TT) { pl = f >> 5; j = f & 31; }
      else           { pl = f / len; j = f - pl * len; }
      v2f v;
      v.x = obuf[pl][j][0];
      v.y = obuf[pl][j][1];
      v2f* dst = (v2f*)out + ((size_t)(pBase + pl) * L_C + (t0 + 1 + j));
      __builtin_nontemporal_store(v, dst);   // write-once stream
    }

    // Next tile's async copies must have landed before the barrier releases.
    wait_async0();
    __syncthreads();
  }
}

extern "C" void kernel_launch(void* const* d_in, const int* in_sizes, int n_in,
                              void* d_out, int out_size, void* d_ws, size_t ws_size,
                              hipStream_t stream) {
  (void)n_in; (void)d_ws; (void)ws_size; (void)out_size;
  const float* x0 = (const float*)d_in[0];
  const float* y0 = (const float*)d_in[1];
  const float* ts = (const float*)d_in[2];
  const float* z  = (const float*)d_in[3];
  float* out = (float*)d_out;

  const int batch  = in_sizes[0];          // 32768
  const int blocks = batch / TPB;          // 512
  sde_path_kernel<<<blocks, TPB, 0, stream>>>(x0, y0, ts, z, out);
}